// DynamicBLTPatcher_71597104825034
// MI455X (gfx1250) — compile-verified
//
#include <hip/hip_runtime.h>
#include <stdint.h>

// ---- CDNA5 vector types ----------------------------------------------------
typedef float        v2f  __attribute__((ext_vector_type(2)));
typedef float        v8f  __attribute__((ext_vector_type(8)));
typedef unsigned int u32x4 __attribute__((ext_vector_type(4)));
typedef int          i32x4 __attribute__((ext_vector_type(4)));
typedef int          i32x8 __attribute__((ext_vector_type(8)));

#define D_MODEL   128
#define VOCAB     256
#define MAX_PATCH 8
#define B_DIM     16
#define T_DIM     16384
// 16 patches (=128 bytes) per wave-group; 8 wave-groups (waves) per block
#define GROUPS_PER_BLOCK 8
#define TOTAL_PATCHES    ((B_DIM * T_DIM) / MAX_PATCH)        // 32768
#define TOTAL_GROUPS     (TOTAL_PATCHES / 16)                 // 2048
#define NUM_BLOCKS       (TOTAL_GROUPS / GROUPS_PER_BLOCK)    // 256

// LDS layout: full embed table (256*128 f32 = 128 KiB) + per-wave index scratch
#define EMB_LDS_FLOATS   (VOCAB * D_MODEL)
#define IDX_LDS_OFF      (EMB_LDS_FLOATS * 4)                 // bytes
#define SMEM_BYTES       (IDX_LDS_OFF + GROUPS_PER_BLOCK * 128 * 4)

__global__ void blt_patch_mean_kernel(const int* __restrict__ x,
                                      const float* __restrict__ emb,
                                      float* __restrict__ out) {
  extern __shared__ char smem[];
  float*        embLds = (float*)smem;
  unsigned int* idxLds = (unsigned int*)(smem + IDX_LDS_OFF);

  const int tid  = threadIdx.x;
  const int wave = tid >> 5;
  const int lane = tid & 31;
  const int half = lane >> 4;   // 0: lanes 0-15, 1: lanes 16-31
  const int n    = lane & 15;   // N index (dim within 16-wide tile)

  // ---- Stage the whole byte_embed table into LDS via the Tensor Data Mover.
  // 2D descriptor: 256 rows x 128 f32, contiguous, pad/iterate/gather off.
  if (tid < 32) {
    uint64_t ga     = (uint64_t)(uintptr_t)emb;
    uint32_t ldsOff = (uint32_t)(uintptr_t)embLds;  // low 32 bits = LDS byte offset
    u32x4 g0;
    g0[0] = 1u;                                     // count=1 (valid), user mode
    g0[1] = ldsOff;                                 // lds_addr
    g0[2] = (uint32_t)ga;                           // global_addr[31:0]
    g0[3] = (uint32_t)((ga >> 32) & 0x1FFFFFFull)   // global_addr[56:32]
          | 0x80000000u;                            // type=2 ("image") at [127:126]
    i32x8 g1;
    g1[0] = (int)(2u << 16);                        // data_size=2 (4B); wg_mask=0
    g1[1] = (int)((uint32_t)D_MODEL << 16);         // tensor_dim0 = 128
    g1[2] = (int)((uint32_t)VOCAB   << 16);         // tensor_dim1 = 256
    g1[3] = (int)((uint32_t)D_MODEL << 16);         // tile_dim0   = 128
    g1[4] = (int)VOCAB;                             // tile_dim1=256, tile_dim2=0
    g1[5] = (int)D_MODEL;                           // tensor_dim0_stride = 128
    g1[6] = (int)(((uint32_t)(VOCAB * D_MODEL) & 0xFFFFu) << 16); // dim1_stride lo
    g1[7] = (int)((uint32_t)(VOCAB * D_MODEL) >> 16);             // dim1_stride hi
    i32x4 zz = {0, 0, 0, 0};
#if defined(__clang_major__) && (__clang_major__ >= 23)
    i32x8 z8 = {0, 0, 0, 0, 0, 0, 0, 0};
    __builtin_amdgcn_tensor_load_to_lds(g0, g1, zz, zz, z8, 0);
#else
    __builtin_amdgcn_tensor_load_to_lds(g0, g1, zz, zz, 0);
#endif
    __builtin_amdgcn_s_wait_tensorcnt(0);           // TDM done before barrier
  }

  // ---- Stage this wave's 128 byte-indices as row *float* offsets into LDS.
  const int g = blockIdx.x * GROUPS_PER_BLOCK + wave;   // 16-patch group id
  const int* xg = x + g * 128;
  unsigned int* myIdx = idxLds + wave * 128;
#pragma unroll
  for (int k = 0; k < 4; ++k) {
    unsigned int row = (unsigned int)xg[k * 32 + lane];
    myIdx[k * 32 + lane] = row * (unsigned int)D_MODEL;
  }

  __syncthreads();   // table + indices visible to all waves; EXEC all-ones again

  // ---- Patch-mean via V_WMMA_F32_16X16X4_F32 (pure fp32, matches reference).
  // D(16 patches x 16 dims) = S(16xK) x E(Kx16), K over 128 bytes in chunks of 4.
  // A-matrix: lane's M = lane%16; value = 1.0 iff byte chunk j belongs to that
  // patch (all 4 bytes of chunk j are in patch j>>1), identical in both A VGPRs,
  // so the accumulation is invariant to the B operand's K-lane permutation.
  v8f acc[8] = {};   // 8 dim-tiles x 8 C VGPRs (fp32 accumulators)

  for (int j = 0; j < 32; ++j) {
    const unsigned int* ip = myIdx + 4 * j + 2 * half;
    unsigned int r0 = ip[0];                         // row offsets for this half's
    unsigned int r1 = ip[1];                         // two K lanes (ds_load_b64)
    float aval = (n == (j >> 1)) ? 1.0f : 0.0f;      // averaging-matrix entry
    v2f A; A[0] = aval; A[1] = aval;
    const float* p0 = embLds + r0 + n;               // LDS gather bases
    const float* p1 = embLds + r1 + n;
#pragma unroll
    for (int t = 0; t < 8; ++t) {                    // 8 tiles cover 128 dims
      v2f Bv; Bv[0] = p0[t * 16]; Bv[1] = p1[t * 16];
      acc[t] = __builtin_amdgcn_wmma_f32_16x16x4_f32(
          /*neg_a=*/false, A, /*neg_b=*/false, Bv,
          /*c_mod=*/(short)0, acc[t], /*reuse_a=*/false, /*reuse_b=*/false);
    }
  }

  // ---- Scale (sum -> mean) and store. C/D layout: VGPR r holds M=r (lanes
  // 0-15) and M=r+8 (lanes 16-31), N = lane%16 -> contiguous 64B per half-wave.
  float* ob = out + ((size_t)g * 16 + (size_t)half * 8) * D_MODEL + n;
#pragma unroll
  for (int t = 0; t < 8; ++t) {
#pragma unroll
    for (int r = 0; r < 8; ++r) {
      ob[(size_t)r * D_MODEL + t * 16] = acc[t][r] * 0.125f;
    }
  }
}

extern "C" void kernel_launch(void* const* d_in, const int* in_sizes, int n_in,
                              void* d_out, int out_size, void* d_ws, size_t ws_size,
                              hipStream_t stream) {
  const int*   x   = (const int*)d_in[0];     // [16, 16384] int32 bytes
  const float* emb = (const float*)d_in[1];   // [256, 128] f32
  float*       out = (float*)d_out;           // [16, 2048, 128] f32
  (void)in_sizes; (void)n_in; (void)out_size; (void)d_ws; (void)ws_size;

  blt_patch_mean_kernel<<<dim3(NUM_BLOCKS), dim3(256), SMEM_BYTES, stream>>>(
      x, emb, out);
}